// ScaleInvariantAttention_62380105007587
// MI455X (gfx1250) — compile-verified
//
#include <hip/hip_runtime.h>
#include <stdint.h>

// Problem constants from the reference: B=64, K=1024, D=512, SCALE=100.
#define SIA_B 64
#define SIA_K 1024
#define SIA_D 512
#define SIA_SCALE 100.0f

typedef float v2f __attribute__((ext_vector_type(2)));
typedef float v8f __attribute__((ext_vector_type(8)));

// ---------------------------------------------------------------------------
// Kernel 0: zero the per-row packed (score,idx) argmax slots in workspace.
// Packed key = (orderedFloatBits(score) << 32) | ~k  -> atomicMax == argmax
// with lowest-index tie-break (matches jnp.argmax).
// ---------------------------------------------------------------------------
__global__ void sia_init(unsigned long long* __restrict__ best) {
    int t = blockIdx.x * blockDim.x + threadIdx.x;
    if (t < SIA_B) best[t] = 0ull;
}

// silu-with-gate term: act * u * sigmoid(u + act), u = (S/sqrt(D))*q*key.
// sigmoid via v_exp_f32 + single v_rcp_f32 (1-ulp rcp is ample for scoring).
__device__ __forceinline__ float sia_term(float qc, float key) {
    float act = (__builtin_fabsf(key) > 1e-6f) ? 1.0f : 0.0f;
    float u = qc * key;
    float x = u + act;
    float sg = __builtin_amdgcn_rcpf(1.0f + __expf(-x));
    return act * (u * sg);
}

// ---------------------------------------------------------------------------
// Kernel 1: scores + per-row argmax.
// Grid: (B, K/64) = (64, 16) -> 1024 blocks, 8192 waves: enough to put ~8
// waves on every SIMD32 of the chip for TRANS-latency hiding. Block: 256
// threads = 8 waves (wave32); each wave owns 8 consecutive keys; each lane
// covers 16 of the 512 dims as 4x float4 (fully coalesced b128 loads).
// Wave32 butterfly reduction, then one global_atomic_max_u64 per wave (best of its 8 keys).
// ---------------------------------------------------------------------------
__global__ __launch_bounds__(256) void sia_scores(
        const float* __restrict__ q,
        const float* __restrict__ keys,
        unsigned long long* __restrict__ best) {
    const int b    = blockIdx.x;
    const int kgrp = blockIdx.y;              // 0..15
    const int lane = threadIdx.x & 31;
    const int wave = threadIdx.x >> 5;        // 0..7
    const float c = SIA_SCALE * 0.04419417382415922f;  // 100 / sqrt(512)

    // Preload this lane's query slice, pre-scaled. d = lane*4 + 128*j + {0..3}.
    float4 qv[4];
    const float4* q4 = (const float4*)(q + (size_t)b * SIA_D);
#pragma unroll
    for (int j = 0; j < 4; ++j) {
        float4 t = q4[lane + 32 * j];
        qv[j] = make_float4(c * t.x, c * t.y, c * t.z, c * t.w);
    }

    unsigned long long bestPacked = 0ull;
    const int kbase = kgrp * 64 + wave * 8;

    for (int kk = 0; kk < 8; ++kk) {
        const int k = kbase + kk;
        const float* krow = keys + (size_t)k * SIA_D;
        // CDNA5 prefetch (global_prefetch_b8) of the next key row into caches.
        __builtin_prefetch(krow + SIA_D, 0, 0);

        float acc = 0.0f;
        const float4* k4 = (const float4*)krow;
#pragma unroll
        for (int j = 0; j < 4; ++j) {
            float4 kv = k4[lane + 32 * j];
            acc += sia_term(qv[j].x, kv.x);
            acc += sia_term(qv[j].y, kv.y);
            acc += sia_term(qv[j].z, kv.z);
            acc += sia_term(qv[j].w, kv.w);
        }
        // wave32 butterfly: all lanes end with the full dot product
#pragma unroll
        for (int off = 16; off > 0; off >>= 1)
            acc += __shfl_xor(acc, off, 32);

        float score = acc * (1.0f / SIA_SCALE);
        unsigned fb = __float_as_uint(score);
        fb = (fb & 0x80000000u) ? ~fb : (fb | 0x80000000u);  // order-preserving
        unsigned long long packed =
            ((unsigned long long)fb << 32) | (unsigned)(~k);
        if (packed > bestPacked) bestPacked = packed;
    }
    if (lane == 0)
        atomicMax(best + b, bestPacked);      // global_atomic_max_u64
}

// ---------------------------------------------------------------------------
// Kernel 2: exact winner gather as a one-hot f32 GEMM via WMMA.
// out[b0:b0+16, n0:n0+16] = onehot(winner)[16 x K] @ values[K x 16]
// One wave per tile (EXEC all 1s, as WMMA requires). A is built on the fly
// per the ISA 16x4 f32 A-layout; B rows loaded per the 4x16 B-layout.
// Chunks with no winner are skipped via a wave-uniform __any() test, so at
// most 16 v_wmma_f32_16x16x4_f32 issue per tile -- exact in f32 (0*x==0,
// 1*x==x for finite inputs), bit-identical to the reference gather.
// ---------------------------------------------------------------------------
__global__ __launch_bounds__(32) void sia_gather(
        const float* __restrict__ values,
        const unsigned long long* __restrict__ best,
        float* __restrict__ outV,
        int* __restrict__ outIdx) {
    const int lane = threadIdx.x;             // 0..31
    const int m    = lane & 15;               // row within 16-row tile
    const int half = lane >> 4;               // 0: K slots {0,1}, 1: {2,3}
    const int b0   = blockIdx.x * 16;
    const int n0   = blockIdx.y * 16;

    const int myWin = (int)(~(unsigned)(best[b0 + m] & 0xffffffffull));
    if (blockIdx.y == 0 && half == 0)
        outIdx[b0 + m] = myWin;               // raw int32 bits after the floats

    v8f cacc = {0.0f, 0.0f, 0.0f, 0.0f, 0.0f, 0.0f, 0.0f, 0.0f};

    for (int k0 = 0; k0 < SIA_K; k0 += 4) {
        bool inRange = (myWin >= k0) && (myWin < k0 + 4);
        if (!__any(inRange)) continue;        // uniform skip: EXEC stays full

        const int kr = k0 + half * 2;
        v2f a, bm;
        a.x = (myWin == kr)     ? 1.0f : 0.0f;   // A: VGPR0 = K 0/2
        a.y = (myWin == kr + 1) ? 1.0f : 0.0f;   //    VGPR1 = K 1/3
        bm.x = values[(size_t)kr * SIA_D + n0 + m];        // B: row K 0/2
        bm.y = values[(size_t)(kr + 1) * SIA_D + n0 + m];  //    row K 1/3

        // D = A(16x4,f32) * B(4x16,f32) + C  ->  v_wmma_f32_16x16x4_f32
        cacc = __builtin_amdgcn_wmma_f32_16x16x4_f32(
            /*neg_a=*/false, a, /*neg_b=*/false, bm,
            /*c_mod=*/(short)0, cacc, /*reuse_a=*/false, /*reuse_b=*/false);
    }

    // C/D layout: VGPR r -> (M = r + 8*half, N = m)
#pragma unroll
    for (int r = 0; r < 8; ++r)
        outV[(size_t)(b0 + r + half * 8) * SIA_D + n0 + m] = cacc[r];
}

// ---------------------------------------------------------------------------
extern "C" void kernel_launch(void* const* d_in, const int* in_sizes, int n_in,
                              void* d_out, int out_size, void* d_ws, size_t ws_size,
                              hipStream_t stream) {
    const float* q      = (const float*)d_in[0];  // [64, 512]
    const float* keys   = (const float*)d_in[1];  // [1024, 512]
    const float* values = (const float*)d_in[2];  // [1024, 512]

    float* outV   = (float*)d_out;                         // [64, 512] f32
    int*   outIdx = (int*)((float*)d_out + SIA_B * SIA_D); // [64] i32 (flat tail)

    unsigned long long* best = (unsigned long long*)d_ws;  // 64 x u64

    sia_init<<<1, 64, 0, stream>>>(best);
    sia_scores<<<dim3(SIA_B, SIA_K / 64), 256, 0, stream>>>(q, keys, best);
    sia_gather<<<dim3(SIA_B / 16, SIA_D / 16), 32, 0, stream>>>(values, best,
                                                                outV, outIdx);
}